// latent_rot_decode_68178310857226
// MI455X (gfx1250) — compile-verified
//
#include <hip/hip_runtime.h>

typedef __attribute__((ext_vector_type(2))) float v2f;
typedef __attribute__((ext_vector_type(8))) float v8f;

// ---------------------------------------------------------------------------
// GEMM: D = act(W @ X)
//   W: (M, Kd) row-major f32, X: (Kd, N) row-major f32, D: (M, N) row-major.
//   Block = 256 threads = 8 wave32; block tile 128(M) x 64(N).
//   Waves arranged 4(M) x 2(N); each wave computes a 32x32 tile as a 2x2 grid
//   of 16x16 V_WMMA_F32_16X16X4_F32 accumulators.
//   A staged row-major (lds_a[m][k]); B staged TRANSPOSED (lds_bt[n][k]) so
//   both A and B fragments are contiguous 8B ds_load_b64 -> no repacking movs,
//   and consecutive-k frags merge into ds_load_2addr_b64.
//   Pitch 36 floats: 16B-aligned staging, conflict-free fragment reads.
// ---------------------------------------------------------------------------
template <bool RELU>
__global__ __launch_bounds__(256) void gemm_wmma_f32(
    const float* __restrict__ W, const float* __restrict__ X,
    float* __restrict__ D, int M, int Kd, int N)
{
    __shared__ float lds_a [128 * 36];  // A tile: 128 m-rows x 32 k, pitch 36
    __shared__ float lds_bt[ 64 * 36];  // B tile:  64 n-rows x 32 k, pitch 36

    const int tid  = threadIdx.x;
    const int lane = tid & 31;
    const int wave = tid >> 5;
    const int lm   = lane & 15;   // row/col-in-subtile selector
    const int half = lane >> 4;   // 0: K 0..1 / M 0..7 ; 1: K 2..3 / M 8..15

    const int blockN = blockIdx.x * 64;
    const int blockM = blockIdx.y * 128;
    const int tileM  = (wave & 3) * 32;   // 4 waves along M
    const int tileN  = (wave >> 2) * 32;  // 2 waves along N

    v8f acc00 = {}, acc01 = {}, acc10 = {}, acc11 = {};

    for (int kk0 = 0; kk0 < Kd; kk0 += 32) {
        // ---- stage A tile: 128x32 floats = 1024 float4, 4 per thread ----
        {
            int idx = tid;
#pragma unroll
            for (int r = 0; r < 4; ++r) {
                const int row = idx >> 3;     // 0..127
                const int c4  = idx & 7;      // 0..7
                const float4 v =
                    *(const float4*)&W[(size_t)(blockM + row) * Kd + kk0 + c4 * 4];
                *(float4*)&lds_a[row * 36 + c4 * 4] = v;
                idx += 256;
            }
        }
        // ---- stage B tile transposed: 32k x 64n floats ----
        // thread: n-group c4 (4 cols), k-pair kp (2 rows); 2 coalesced b128
        // loads, 4 transposed b64 stores (k-pairs contiguous in lds_bt).
        {
            const int c4 = tid & 15;          // n-group: n = c4*4 .. c4*4+3
            const int kp = tid >> 4;          // k-pair:  k = 2*kp, 2*kp+1
            const float4 vlo =
                *(const float4*)&X[(size_t)(kk0 + 2 * kp    ) * N + blockN + c4 * 4];
            const float4 vhi =
                *(const float4*)&X[(size_t)(kk0 + 2 * kp + 1) * N + blockN + c4 * 4];
            v2f s;
            s.x = vlo.x; s.y = vhi.x;
            *(v2f*)&lds_bt[(c4 * 4 + 0) * 36 + 2 * kp] = s;
            s.x = vlo.y; s.y = vhi.y;
            *(v2f*)&lds_bt[(c4 * 4 + 1) * 36 + 2 * kp] = s;
            s.x = vlo.z; s.y = vhi.z;
            *(v2f*)&lds_bt[(c4 * 4 + 2) * 36 + 2 * kp] = s;
            s.x = vlo.w; s.y = vhi.w;
            *(v2f*)&lds_bt[(c4 * 4 + 3) * 36 + 2 * kp] = s;
        }
        __syncthreads();

        // ---- 8 K-steps of 4; 4 WMMAs per step (2x2 outer product) ----
#pragma unroll
        for (int kk = 0; kk < 32; kk += 4) {
            // frags: lanes 0-15 -> K=kk..kk+1, lanes 16-31 -> K=kk+2..kk+3
            v2f a0 = *(const v2f*)&lds_a [(tileM +      lm) * 36 + kk + half * 2];
            v2f a1 = *(const v2f*)&lds_a [(tileM + 16 + lm) * 36 + kk + half * 2];
            v2f b0 = *(const v2f*)&lds_bt[(tileN +      lm) * 36 + kk + half * 2];
            v2f b1 = *(const v2f*)&lds_bt[(tileN + 16 + lm) * 36 + kk + half * 2];

            acc00 = __builtin_amdgcn_wmma_f32_16x16x4_f32(
                false, a0, false, b0, (short)0, acc00, false, false);
            acc01 = __builtin_amdgcn_wmma_f32_16x16x4_f32(
                false, a0, false, b1, (short)0, acc01, false, false);
            acc10 = __builtin_amdgcn_wmma_f32_16x16x4_f32(
                false, a1, false, b0, (short)0, acc10, false, false);
            acc11 = __builtin_amdgcn_wmma_f32_16x16x4_f32(
                false, a1, false, b1, (short)0, acc11, false, false);
        }
        __syncthreads();
    }

    // ---- write back 4 subtiles: VGPR i -> M = i + half*8, N = lm ----
    const int col0 = blockN + tileN + lm;
    const int row0 = blockM + tileM + half * 8;
#pragma unroll
    for (int i = 0; i < 8; ++i) {
        float v00 = acc00[i], v01 = acc01[i], v10 = acc10[i], v11 = acc11[i];
        if (RELU) {
            v00 = fmaxf(v00, 0.0f); v01 = fmaxf(v01, 0.0f);
            v10 = fmaxf(v10, 0.0f); v11 = fmaxf(v11, 0.0f);
        }
        D[(size_t)(row0 + i     ) * N + col0     ] = v00;
        D[(size_t)(row0 + i     ) * N + col0 + 16] = v01;
        D[(size_t)(row0 + i + 16) * N + col0     ] = v10;
        D[(size_t)(row0 + i + 16) * N + col0 + 16] = v11;
    }
}

// ---------------------------------------------------------------------------
// 3x3 matrix exponential via scaling-and-squaring + order-10 Horner Taylor.
// One thread per (part, column). Writes rotations region and the rotation
// rows (a*3+c, a<3) of the 12-row-per-part transform tensor.
// K must be a power of two.
// ---------------------------------------------------------------------------
__device__ __forceinline__ void mat3mul(float* __restrict__ o,
                                        const float* __restrict__ x,
                                        const float* __restrict__ y)
{
#pragma unroll
    for (int i = 0; i < 3; ++i)
#pragma unroll
        for (int j = 0; j < 3; ++j)
            o[i * 3 + j] = x[i * 3 + 0] * y[0 * 3 + j] +
                           x[i * 3 + 1] * y[1 * 3 + j] +
                           x[i * 3 + 2] * y[2 * 3 + j];
}

__global__ __launch_bounds__(256) void expm3x3_kernel(
    const float* __restrict__ omega, float* __restrict__ rot,
    float* __restrict__ transf, int parts, int K)
{
    const int idx   = blockIdx.x * blockDim.x + threadIdx.x;
    const int total = parts * K;
    if (idx >= total) return;
    const int col = idx & (K - 1);
    const int p   = idx >> (31 - __clz(K));   // idx / K, K power of two

    float a[9];
#pragma unroll
    for (int i = 0; i < 9; ++i)
        a[i] = omega[(size_t)(9 * p + i) * K + col];

    // inf-norm (max abs row sum)
    const float n0 = fabsf(a[0]) + fabsf(a[1]) + fabsf(a[2]);
    const float n1 = fabsf(a[3]) + fabsf(a[4]) + fabsf(a[5]);
    const float n2 = fabsf(a[6]) + fabsf(a[7]) + fabsf(a[8]);
    const float nrm = fmaxf(n0, fmaxf(n1, n2));

    int s = 0;
    if (nrm > 0.25f) {
        s = (int)ceilf(log2f(nrm * 4.0f));
        if (s < 0)  s = 0;
        if (s > 40) s = 40;
    }
    const float sc = exp2f((float)-s);
#pragma unroll
    for (int i = 0; i < 9; ++i) a[i] *= sc;

    // R = I + B(I + B/2(I + B/3(... (I + B/10) ...)))
    float r[9], t[9];
#pragma unroll
    for (int i = 0; i < 9; ++i) r[i] = (i % 4 == 0) ? 1.0f : 0.0f;
    for (int j = 10; j >= 1; --j) {
        const float inv = 1.0f / (float)j;
        mat3mul(t, a, r);
#pragma unroll
        for (int i = 0; i < 9; ++i)
            r[i] = ((i % 4 == 0) ? 1.0f : 0.0f) + t[i] * inv;
    }
    // undo scaling: square s times
    for (int q = 0; q < s; ++q) {
        mat3mul(t, r, r);
#pragma unroll
        for (int i = 0; i < 9; ++i) r[i] = t[i];
    }

#pragma unroll
    for (int i = 0; i < 9; ++i) {
        rot   [(size_t)(9  * p + i) * K + col] = r[i];
        transf[(size_t)(12 * p + i) * K + col] = r[i];  // rows a*3+c, a<3
    }
}

// ---------------------------------------------------------------------------
// Pack translations into the 4th row-group of each 12-row transform block:
// transf[p*12 + 9 + c] = translations[p*3 + c]
// ---------------------------------------------------------------------------
__global__ __launch_bounds__(256) void pack_translations(
    const float* __restrict__ trans, float* __restrict__ transf,
    int parts, int K)
{
    const int idx   = blockIdx.x * blockDim.x + threadIdx.x;
    const int total = parts * 3 * K;
    if (idx >= total) return;
    const int col = idx & (K - 1);
    const int r   = idx >> (31 - __clz(K));   // idx / K, K power of two
    const int p   = r / 3;
    const int c   = r % 3;
    transf[(size_t)(p * 12 + 9 + c) * K + col] = trans[(size_t)r * K + col];
}

// ---------------------------------------------------------------------------
extern "C" void kernel_launch(void* const* d_in, const int* in_sizes, int n_in,
                              void* d_out, int out_size, void* d_ws, size_t ws_size,
                              hipStream_t stream)
{
    (void)in_sizes; (void)n_in; (void)out_size; (void)ws_size;

    const float* x   = (const float*)d_in[0];
    const float* Wo0 = (const float*)d_in[1];
    const float* Wo1 = (const float*)d_in[2];
    const float* Wo2 = (const float*)d_in[3];
    const float* Wo3 = (const float*)d_in[4];
    const float* Wt0 = (const float*)d_in[5];
    const float* Wt1 = (const float*)d_in[6];
    const float* Wt2 = (const float*)d_in[7];
    const float* Wt3 = (const float*)d_in[8];

    const int K = 2048, LAT = 128, HID = 2048, P = 128;
    const int OD = 9 * P;    // 1152
    const int TD = 3 * P;    // 384
    const int XD = 12 * P;   // 1536

    float* out    = (float*)d_out;
    float* omega  = out;                             // (1152, 2048)
    float* transf = omega  + (size_t)OD * K;         // (1536, 2048)
    float* rot    = transf + (size_t)XD * K;         // (1152, 2048)
    float* trans  = rot    + (size_t)OD * K;         // ( 384, 2048)

    float* h0 = (float*)d_ws;                        // (2048, 2048)
    float* h1 = h0 + (size_t)HID * K;                // (2048, 2048)

    const dim3 blk(256);
    auto gemm = [&](const float* W, const float* Xin, float* Dst,
                    int M, int Kd, bool relu) {
        dim3 grid(K / 64, M / 128);
        if (relu)
            gemm_wmma_f32<true ><<<grid, blk, 0, stream>>>(W, Xin, Dst, M, Kd, K);
        else
            gemm_wmma_f32<false><<<grid, blk, 0, stream>>>(W, Xin, Dst, M, Kd, K);
    };

    // omega branch
    gemm(Wo0, x,  h0,    HID, LAT, true);
    gemm(Wo1, h0, h1,    HID, HID, true);
    gemm(Wo2, h1, h0,    HID, HID, true);
    gemm(Wo3, h0, omega, OD,  HID, false);

    // translation branch (reuses h0/h1 after omega branch is done)
    gemm(Wt0, x,  h1,    HID, LAT, true);
    gemm(Wt1, h1, h0,    HID, HID, true);
    gemm(Wt2, h0, h1,    HID, HID, true);
    gemm(Wt3, h1, trans, TD,  HID, false);

    // expm over all (part, column) 3x3 blocks -> rotations + transf rows
    {
        const int total = P * K;
        expm3x3_kernel<<<(total + 255) / 256, 256, 0, stream>>>(omega, rot, transf, P, K);
    }
    // drop translations into transf 4th row-group
    {
        const int total = TD * K;
        pack_translations<<<(total + 255) / 256, 256, 0, stream>>>(trans, transf, P, K);
    }
}